// L2ECELoss_79809082294501
// MI455X (gfx1250) — compile-verified
//
#include <hip/hip_runtime.h>

#define THREADS 256
#define WAVES (THREADS / 32)
#define NBINS 15

typedef __attribute__((ext_vector_type(2))) float v2f;
typedef __attribute__((ext_vector_type(8))) float v8f;
typedef int v4i __attribute__((vector_size(16)));
typedef __attribute__((address_space(1))) v4i* gp_v4i;
typedef __attribute__((address_space(3))) v4i* lp_v4i;

#if defined(__AMDGCN__) && defined(__has_builtin)
#if __has_builtin(__builtin_amdgcn_global_load_async_to_lds_b128) && \
    __has_builtin(__builtin_amdgcn_s_wait_asynccnt)
#define HAVE_ASYNC 1
#endif
#if __has_builtin(__builtin_amdgcn_wmma_f32_16x16x4_f32)
#define HAVE_WMMA4 1
#endif
#endif
#ifndef HAVE_ASYNC
#define HAVE_ASYNC 0
#endif
#ifndef HAVE_WMMA4
#define HAVE_WMMA4 0
#endif

// ---------------------------------------------------------------- zero ws
__global__ void ece_zero(float* ws) {
    int t = threadIdx.x;
    if (t < 48) ws[t] = 0.0f;
}

// ------------------------------------------------------------- accumulate
__device__ __forceinline__ void ece_acc1(float c, float a, float (*h)[16]) {
    // reference: bin b covers (b/15, (b+1)/15]; invalid if c<=0 or c>1
    float w = (c > 0.0f && c <= 1.0f) ? 1.0f : 0.0f;
    int b = (int)ceilf(c * 15.0f) - 1;
    b = b < 0 ? 0 : (b > NBINS - 1 ? NBINS - 1 : b);
    atomicAdd(&h[0][b], w);          // count
    atomicAdd(&h[1][b], c * w);      // conf sum
    atomicAdd(&h[2][b], a * w);      // acc  sum
}

__global__ void __launch_bounds__(THREADS)
ece_histo(const float* __restrict__ conf, const float* __restrict__ acc,
          float* __restrict__ ws, int n) {
    __shared__ float hist[WAVES][3][16];
    const int wave = threadIdx.x >> 5;
    float (*h)[16] = hist[wave];

    for (int i = threadIdx.x; i < WAVES * 48; i += THREADS)
        ((float*)hist)[i] = 0.0f;
    __syncthreads();

    const int gtid = blockIdx.x * blockDim.x + threadIdx.x;
    const int stride = gridDim.x * blockDim.x;
    const int n4 = n >> 2;
    const float4* c4p = (const float4*)conf;
    const float4* a4p = (const float4*)acc;

#if HAVE_ASYNC
    __shared__ float4 sC[2][THREADS];
    __shared__ float4 sA[2][THREADS];
    const int nt = n4 / stride;  // uniform trip count (N = 2^25 divides evenly)
    if (nt > 0) {
        // prologue: stage tile 0 (each lane stages its own 16B, no barrier needed)
        __builtin_amdgcn_global_load_async_to_lds_b128(
            (gp_v4i)(c4p + gtid), (lp_v4i)&sC[0][threadIdx.x], 0, 0);
        __builtin_amdgcn_global_load_async_to_lds_b128(
            (gp_v4i)(a4p + gtid), (lp_v4i)&sA[0][threadIdx.x], 0, 0);
        for (int t = 0; t < nt; ++t) {
            if (t + 1 < nt) {
                const int idx = (t + 1) * stride + gtid;
                __builtin_amdgcn_global_load_async_to_lds_b128(
                    (gp_v4i)(c4p + idx), (lp_v4i)&sC[(t + 1) & 1][threadIdx.x], 0, 0);
                __builtin_amdgcn_global_load_async_to_lds_b128(
                    (gp_v4i)(a4p + idx), (lp_v4i)&sA[(t + 1) & 1][threadIdx.x], 0, 0);
                __builtin_amdgcn_s_wait_asynccnt(2);  // tile t's pair complete (in-order)
            } else {
                __builtin_amdgcn_s_wait_asynccnt(0);
            }
            float4 c = sC[t & 1][threadIdx.x];
            float4 a = sA[t & 1][threadIdx.x];
            ece_acc1(c.x, a.x, h);
            ece_acc1(c.y, a.y, h);
            ece_acc1(c.z, a.z, h);
            ece_acc1(c.w, a.w, h);
        }
    }
    // generic tail (empty for N = 2^25)
    for (int i = nt * stride + gtid; i < n4; i += stride) {
        float4 c = c4p[i], a = a4p[i];
        ece_acc1(c.x, a.x, h);
        ece_acc1(c.y, a.y, h);
        ece_acc1(c.z, a.z, h);
        ece_acc1(c.w, a.w, h);
    }
#else
    for (int i = gtid; i < n4; i += stride) {
        float4 c = c4p[i], a = a4p[i];
        ece_acc1(c.x, a.x, h);
        ece_acc1(c.y, a.y, h);
        ece_acc1(c.z, a.z, h);
        ece_acc1(c.w, a.w, h);
    }
#endif

    __syncthreads();
    // cross-wave reduce, then 48 global f32 atomics per block
    for (int i = threadIdx.x; i < 48; i += THREADS) {
        float s = 0.0f;
        for (int w = 0; w < WAVES; ++w) s += hist[w][i >> 4][i & 15];
        atomicAdd(&ws[i], s);
    }
}

// --------------------------------------------------------------- finalize
// 1 wave of 32 threads. Lanes 0..14 compute per-bin terms; the 16-term sum is
// done exactly in f32 on the matrix unit: two chained V_WMMA_F32_16X16X4_F32
// with A = ones (C[m][n] = sum_k B[k][n]) and B gathered via ds_bpermute.
__global__ void ece_finalize(const float* __restrict__ ws, float* __restrict__ out,
                             float inv_n) {
    const int lane = threadIdx.x;  // 0..31, wave32
    float p = 0.0f;
    if (lane < NBINS) {
        float cnt = ws[lane];
        float cs = ws[16 + lane];
        float as_ = ws[32 + lane];
        float safe = fmaxf(cnt, 1.0f);
        float d = cs / safe - as_ / safe;
        p = (cnt > 0.0f) ? d * d * cnt * inv_n : 0.0f;
    }
#if HAVE_WMMA4
    const int n16 = lane & 15;
    const int k2 = lane >> 4;  // which K-pair this lane holds (0 -> K0/K1, 1 -> K2/K3)
    const int pi = __float_as_int(p);
    v2f A;
    A.x = 1.0f;
    A.y = 1.0f;
    // Stage 1: B[0..3][n] = p[4n], p[4n+2], p[4n+1], p[4n+3]  (p == 0 for source lanes >= 15; columns n>=8
    // wrap in bpermute but are never consumed by stage 2)
    v2f B;
    B.x = __int_as_float(__builtin_amdgcn_ds_bpermute((4 * n16 + k2) * 4, pi));
    B.y = __int_as_float(__builtin_amdgcn_ds_bpermute((4 * n16 + 2 + k2) * 4, pi));
    v8f C = {};
    C = __builtin_amdgcn_wmma_f32_16x16x4_f32(false, A, false, B, (short)0, C,
                                              false, false);
    // C row 0: lane n (n<16) holds s_n = p[4n]+p[4n+1]+p[4n+2]+p[4n+3]
    const int si = __float_as_int(C[0]);
    // Stage 2: B2[0..3][n] = s_0, s_2, s_1, s_3 for every n  ->  C2[0][n] = s_0+s_1+s_2+s_3
    v2f B2;
    B2.x = __int_as_float(__builtin_amdgcn_ds_bpermute((k2)*4, si));
    B2.y = __int_as_float(__builtin_amdgcn_ds_bpermute((k2 + 2) * 4, si));
    v8f C2 = {};
    C2 = __builtin_amdgcn_wmma_f32_16x16x4_f32(false, A, false, B2, (short)0, C2,
                                               false, false);
    if (lane == 0) out[0] = C2[0];
#else
    __shared__ float sp[32];
    sp[lane] = p;
    __syncthreads();
    if (lane == 0) {
        float s = 0.0f;
        for (int i = 0; i < NBINS; ++i) s += sp[i];
        out[0] = s;
    }
#endif
}

// ----------------------------------------------------------------- launch
extern "C" void kernel_launch(void* const* d_in, const int* in_sizes, int n_in,
                              void* d_out, int out_size, void* d_ws, size_t ws_size,
                              hipStream_t stream) {
    const float* conf = (const float*)d_in[0];
    const float* acc = (const float*)d_in[1];
    float* ws = (float*)d_ws;
    float* out = (float*)d_out;
    const int n = in_sizes[0];

    ece_zero<<<1, 64, 0, stream>>>(ws);

    int n4 = n >> 2;
    int blocks = (n4 + THREADS - 1) / THREADS;
    if (blocks > 4096) blocks = 4096;
    if (blocks < 1) blocks = 1;
    ece_histo<<<blocks, THREADS, 0, stream>>>(conf, acc, ws, n);

    ece_finalize<<<1, 32, 0, stream>>>(ws, out, 1.0f / (float)n);
}